// RNN_38268158607516
// MI455X (gfx1250) — compile-verified
//
#include <hip/hip_runtime.h>
#include <math.h>

typedef float v2f __attribute__((ext_vector_type(2)));
typedef float v8f __attribute__((ext_vector_type(8)));

#define Bb_ 4096
#define Tt_ 512
#define LST 18   // LDS row stride (floats): keeps b64 loads 8B-aligned

__device__ __forceinline__ v8f wmma4(v2f a, v2f b, v8f c) {
  // V_WMMA_F32_16X16X4_F32 : D = A(16x4) x B(4x16) + C(16x16), fp32
  return __builtin_amdgcn_wmma_f32_16x16x4_f32(
      false, a, false, b, (short)0, c, false, false);
}

// Branch-free tanh: v_tanh_f32 if available, else v_exp_f32 + v_rcp_f32.
__device__ __forceinline__ float fast_tanh(float x) {
#if __has_builtin(__builtin_amdgcn_tanhf)
  return __builtin_amdgcn_tanhf(x);
#else
  // tanh(x) = 1 - 2/(exp2(2*log2e*x) + 1); saturates correctly at +-inf.
  float e = __builtin_amdgcn_exp2f(x * 2.8853900817779268f);
  return 1.0f - 2.0f * __builtin_amdgcn_rcpf(e + 1.0f);
#endif
}

// ---------------------------------------------------------------------------
// Layers 1..3: Din = 16 (2H). seqIn/seqOut: [T][B][16] fp32.
// One wave = 16 batch rows, both directions packed into one 16x16 C tile.
// Single-wave workgroup: intra-wave LDS is in-order -> no barriers needed.
// ---------------------------------------------------------------------------
__global__ __launch_bounds__(32) void rnn_layer16(
    const float* __restrict__ seqIn, float* __restrict__ seqOut,
    const float* __restrict__ Wih,   // [2][8][16]
    const float* __restrict__ Whh,   // [2][8][8]
    const float* __restrict__ bih,   // [2][8]
    const float* __restrict__ bhh) { // [2][8]
  __shared__ float hbuf[16 * LST];
  const int l    = threadIdx.x;
  const int n    = l & 15;                 // N column (and A-row m)
  const int kofs = (l < 16) ? 0 : 2;       // A/B lane K split
  const int bt   = blockIdx.x;

  // Build B operands in registers (once).
  v2f Bf[4], Bbk[4], Br[4];
#pragma unroll
  for (int c = 0; c < 4; ++c) {
    int k0 = 4 * c + kofs;
    Bf[c].x  = (n <  8) ? Wih[n * 16 + k0]                 : 0.f;
    Bf[c].y  = (n <  8) ? Wih[n * 16 + k0 + 1]             : 0.f;
    Bbk[c].x = (n >= 8) ? Wih[128 + (n - 8) * 16 + k0]     : 0.f;
    Bbk[c].y = (n >= 8) ? Wih[128 + (n - 8) * 16 + k0 + 1] : 0.f;
    float r0 = 0.f, r1 = 0.f;
    if (n < 8 && k0 < 8)            r0 = Whh[n * 8 + k0];
    else if (n >= 8 && k0 >= 8)     r0 = Whh[64 + (n - 8) * 8 + (k0 - 8)];
    int k1 = k0 + 1;
    if (n < 8 && k1 < 8)            r1 = Whh[n * 8 + k1];
    else if (n >= 8 && k1 >= 8)     r1 = Whh[64 + (n - 8) * 8 + (k1 - 8)];
    Br[c].x = r0; Br[c].y = r1;
  }
  const float bias = bih[n] + bhh[n];      // [2][8] flat: n indexes both dirs

  for (int i = l; i < 16 * LST; i += 32) hbuf[i] = 0.f;  // h0 = 0
  __syncthreads();

  const int    m       = n;                            // batch row in tile
  const size_t rowBase = (size_t)(bt * 16 + m) * 16;
  const int    Mb      = (l < 16) ? 0 : 8;

  // Software pipeline: preload t=0 projection operands.
  v2f af[4], ab[4];
  {
    const float* pf0 = seqIn + rowBase;
    const float* pb0 = seqIn + (size_t)(Tt_ - 1) * Bb_ * 16 + rowBase;
#pragma unroll
    for (int c = 0; c < 4; ++c) {
      af[c] = *(const v2f*)(pf0 + 4 * c + kofs);
      ab[c] = *(const v2f*)(pb0 + 4 * c + kofs);
    }
  }

  for (int t = 0; t < Tt_; ++t) {
    // (1) recurrence operands from LDS, issued first (latency overlaps proj)
    v2f ah[4];
#pragma unroll
    for (int c = 0; c < 4; ++c)
      ah[c] = *(const v2f*)(&hbuf[m * LST + 4 * c + kofs]);

    // (2) issue next timestep's projection loads (hidden behind this step)
    v2f naf[4], nab[4];
    if (t + 1 < Tt_) {
      const float* pf = seqIn + (size_t)(t + 1) * Bb_ * 16 + rowBase;
      const float* pb = seqIn + (size_t)(Tt_ - 2 - t) * Bb_ * 16 + rowBase;
      __builtin_prefetch(pf + (size_t)Bb_ * 16, 0, 1);
      __builtin_prefetch(pb - (size_t)Bb_ * 16, 0, 1);
#pragma unroll
      for (int c = 0; c < 4; ++c) {
        naf[c] = *(const v2f*)(pf + 4 * c + kofs);
        nab[c] = *(const v2f*)(pb + 4 * c + kofs);
      }
    } else {
#pragma unroll
      for (int c = 0; c < 4; ++c) { naf[c] = af[c]; nab[c] = ab[c]; }
    }

    // (3) projection chain (independent of h)
    v8f accP;
#pragma unroll
    for (int r = 0; r < 8; ++r) accP[r] = bias;
#pragma unroll
    for (int c = 0; c < 4; ++c) accP = wmma4(af[c], Bf[c], accP);
#pragma unroll
    for (int c = 0; c < 4; ++c) accP = wmma4(ab[c], Bbk[c], accP);

    // (4) recurrence chain (critical path: only 4 WMMAs deep)
    v8f accR = {};
#pragma unroll
    for (int c = 0; c < 4; ++c) accR = wmma4(ah[c], Br[c], accR);

    // (5) combine + activation
#pragma unroll
    for (int r = 0; r < 8; ++r) accR[r] = fast_tanh(accP[r] + accR[r]);

    // (6) write h (LDS, in-order within wave) + layer output (global)
    float* outp = (n < 8) ? (seqOut + (size_t)t * Bb_ * 16)
                          : (seqOut + (size_t)(Tt_ - 1 - t) * Bb_ * 16);
#pragma unroll
    for (int r = 0; r < 8; ++r) {
      int M = Mb + r;
      hbuf[M * LST + n] = accR[r];
      outp[(size_t)(bt * 16 + M) * 16 + n] = accR[r];
    }

#pragma unroll
    for (int c = 0; c < 4; ++c) { af[c] = naf[c]; ab[c] = nab[c]; }
  }
}

// ---------------------------------------------------------------------------
// Layer 0: Din = 6, input x is [B][6][T] (feature-major over time).
// ---------------------------------------------------------------------------
__global__ __launch_bounds__(32) void rnn_layer0(
    const float* __restrict__ x, float* __restrict__ seqOut,
    const float* __restrict__ Wih0,  // [2][8][6]
    const float* __restrict__ Whh0,  // [2][8][8]
    const float* __restrict__ bih,   // [2][8]
    const float* __restrict__ bhh) { // [2][8]
  __shared__ float hbuf[16 * LST];
  const int l    = threadIdx.x;
  const int n    = l & 15;
  const int kofs = (l < 16) ? 0 : 2;
  const int bt   = blockIdx.x;

  v2f Bf[2], Bbk[2], Br[4];
#pragma unroll
  for (int c = 0; c < 2; ++c) {
    int k0 = 4 * c + kofs;
    Bf[c].x  = (n <  8 && k0     < 6) ? Wih0[n * 6 + k0]                : 0.f;
    Bf[c].y  = (n <  8 && k0 + 1 < 6) ? Wih0[n * 6 + k0 + 1]            : 0.f;
    Bbk[c].x = (n >= 8 && k0     < 6) ? Wih0[48 + (n - 8) * 6 + k0]     : 0.f;
    Bbk[c].y = (n >= 8 && k0 + 1 < 6) ? Wih0[48 + (n - 8) * 6 + k0 + 1] : 0.f;
  }
#pragma unroll
  for (int c = 0; c < 4; ++c) {
    int k0 = 4 * c + kofs;
    float r0 = 0.f, r1 = 0.f;
    if (n < 8 && k0 < 8)        r0 = Whh0[n * 8 + k0];
    else if (n >= 8 && k0 >= 8) r0 = Whh0[64 + (n - 8) * 8 + (k0 - 8)];
    int k1 = k0 + 1;
    if (n < 8 && k1 < 8)        r1 = Whh0[n * 8 + k1];
    else if (n >= 8 && k1 >= 8) r1 = Whh0[64 + (n - 8) * 8 + (k1 - 8)];
    Br[c].x = r0; Br[c].y = r1;
  }
  const float bias = bih[n] + bhh[n];

  for (int i = l; i < 16 * LST; i += 32) hbuf[i] = 0.f;
  __syncthreads();

  const int m  = n;
  const int Mb = (l < 16) ? 0 : 8;
  const float* px = x + (size_t)(bt * 16 + m) * 6 * Tt_;

  // Preload t=0 operands (features strided by T in memory).
  v2f af[2], ab[2];
#pragma unroll
  for (int c = 0; c < 2; ++c) {
    int k0 = 4 * c + kofs;
    af[c].x = (k0     < 6) ? px[k0 * Tt_]              : 0.f;
    af[c].y = (k0 + 1 < 6) ? px[(k0 + 1) * Tt_]        : 0.f;
    ab[c].x = (k0     < 6) ? px[k0 * Tt_ + Tt_ - 1]    : 0.f;
    ab[c].y = (k0 + 1 < 6) ? px[(k0 + 1) * Tt_ + Tt_ - 1] : 0.f;
  }

  for (int t = 0; t < Tt_; ++t) {
    v2f ah[4];
#pragma unroll
    for (int c = 0; c < 4; ++c)
      ah[c] = *(const v2f*)(&hbuf[m * LST + 4 * c + kofs]);

    v2f naf[2], nab[2];
    if (t + 1 < Tt_) {
      const int tn = t + 1, tbn = Tt_ - 2 - t;
#pragma unroll
      for (int c = 0; c < 2; ++c) {
        int k0 = 4 * c + kofs;
        naf[c].x = (k0     < 6) ? px[k0 * Tt_ + tn]        : 0.f;
        naf[c].y = (k0 + 1 < 6) ? px[(k0 + 1) * Tt_ + tn]  : 0.f;
        nab[c].x = (k0     < 6) ? px[k0 * Tt_ + tbn]       : 0.f;
        nab[c].y = (k0 + 1 < 6) ? px[(k0 + 1) * Tt_ + tbn] : 0.f;
      }
    } else {
#pragma unroll
      for (int c = 0; c < 2; ++c) { naf[c] = af[c]; nab[c] = ab[c]; }
    }

    v8f accP;
#pragma unroll
    for (int r = 0; r < 8; ++r) accP[r] = bias;
#pragma unroll
    for (int c = 0; c < 2; ++c) {
      accP = wmma4(af[c], Bf[c], accP);
      accP = wmma4(ab[c], Bbk[c], accP);
    }

    v8f accR = {};
#pragma unroll
    for (int c = 0; c < 4; ++c) accR = wmma4(ah[c], Br[c], accR);

#pragma unroll
    for (int r = 0; r < 8; ++r) accR[r] = fast_tanh(accP[r] + accR[r]);

    float* outp = (n < 8) ? (seqOut + (size_t)t * Bb_ * 16)
                          : (seqOut + (size_t)(Tt_ - 1 - t) * Bb_ * 16);
#pragma unroll
    for (int r = 0; r < 8; ++r) {
      int M = Mb + r;
      hbuf[M * LST + n] = accR[r];
      outp[(size_t)(bt * 16 + M) * 16 + n] = accR[r];
    }

#pragma unroll
    for (int c = 0; c < 2; ++c) { af[c] = naf[c]; ab[c] = nab[c]; }
  }
}

// ---------------------------------------------------------------------------
// Final FC: out[b][c] = seq[T-1][b][:] . fc_w[c][:] + fc_b[c]
// ---------------------------------------------------------------------------
__global__ void fc_kernel(const float* __restrict__ seq,
                          const float* __restrict__ fcw,
                          const float* __restrict__ fcb,
                          float* __restrict__ out) {
  int b = blockIdx.x * blockDim.x + threadIdx.x;
  if (b >= Bb_) return;
  const float* h = seq + (size_t)(Tt_ - 1) * Bb_ * 16 + (size_t)b * 16;
  float a0 = fcb[0], a1 = fcb[1];
#pragma unroll
  for (int k = 0; k < 16; ++k) {
    float v = h[k];
    a0 = fmaf(v, fcw[k], a0);
    a1 = fmaf(v, fcw[16 + k], a1);
  }
  out[b * 2 + 0] = a0;
  out[b * 2 + 1] = a1;
}

extern "C" void kernel_launch(void* const* d_in, const int* in_sizes, int n_in,
                              void* d_out, int out_size, void* d_ws, size_t ws_size,
                              hipStream_t stream) {
  const float* x    = (const float*)d_in[0];
  const float* Wih0 = (const float*)d_in[1];
  const float* Whh0 = (const float*)d_in[2];
  const float* bih0 = (const float*)d_in[3];
  const float* bhh0 = (const float*)d_in[4];
  const float* Wih  = (const float*)d_in[5];   // [3][2][8][16]
  const float* Whh  = (const float*)d_in[6];   // [3][2][8][8]
  const float* bih  = (const float*)d_in[7];   // [3][2][8]
  const float* bhh  = (const float*)d_in[8];
  const float* fcw  = (const float*)d_in[9];
  const float* fcb  = (const float*)d_in[10];
  float* out = (float*)d_out;

  float* seqA = (float*)d_ws;                       // [T][B][16] = 128 MB
  float* seqB = seqA + (size_t)Tt_ * Bb_ * 16;      // second ping-pong buffer

  dim3 grid(Bb_ / 16), block(32);
  rnn_layer0 <<<grid, block, 0, stream>>>(x,    seqA, Wih0, Whh0, bih0, bhh0);
  rnn_layer16<<<grid, block, 0, stream>>>(seqA, seqB, Wih + 0 * 256, Whh + 0 * 128,
                                          bih + 0 * 16, bhh + 0 * 16);
  rnn_layer16<<<grid, block, 0, stream>>>(seqB, seqA, Wih + 1 * 256, Whh + 1 * 128,
                                          bih + 1 * 16, bhh + 1 * 16);
  rnn_layer16<<<grid, block, 0, stream>>>(seqA, seqB, Wih + 2 * 256, Whh + 2 * 128,
                                          bih + 2 * 16, bhh + 2 * 16);
  fc_kernel<<<dim3((Bb_ + 255) / 256), dim3(256), 0, stream>>>(seqB, fcw, fcb, out);
}